// Attention_50757923504387
// MI455X (gfx1250) — compile-verified
//
#include <hip/hip_runtime.h>
#include <hip/hip_bf16.h>

// Problem sizes (fixed by reference)
#define BATCH 32
#define SLEN  2048
#define DIM   1024
#define SPLIT 16            // S-splits per batch for the flash pass

typedef __attribute__((ext_vector_type(2))) float v2f;
typedef __attribute__((ext_vector_type(8))) float v8f;

// ---------------------------------------------------------------------------
// Kernel 1: u[b,d] = sum_e st[b,e] * W[e,d]   (M=32, N=1024, K=1024)
// fp32 WMMA 16x16x4. 2 M-tiles x 64 N-tiles = 128 waves -> 16 blocks x 256.
// A 16x4 layout: lanes 0-15 hold M=0..15; VGPR0 = K={0|2}, VGPR1 = K={1|3}
// (lane>=16 -> K base 2). B 4x16 mirrored: lanes hold N, VGPRs hold K pairs.
// C/D: VGPR i -> M = i + 8*(lane>=16), N = lane%16.
// ---------------------------------------------------------------------------
__global__ __launch_bounds__(256) void k1_u_gemm(const float* __restrict__ st,
                                                 const float* __restrict__ W,
                                                 float* __restrict__ u) {
  const int wave = (blockIdx.x * blockDim.x + threadIdx.x) >> 5;
  const int lane = threadIdx.x & 31;
  const int mt = wave >> 6;          // 0..1
  const int nt = wave & 63;          // 0..63
  const int row = lane & 15;
  const int kh  = lane >> 4;         // 0 or 1
  const int m0 = mt * 16, n0 = nt * 16;

  v8f acc = {};
  for (int k = 0; k < DIM; k += 4) {
    const int kk = k + 2 * kh;
    v2f a, bm;
    a.x  = st[(m0 + row) * DIM + kk];
    a.y  = st[(m0 + row) * DIM + kk + 1];
    bm.x = W[(size_t)kk * DIM + n0 + row];
    bm.y = W[(size_t)(kk + 1) * DIM + n0 + row];
    acc = __builtin_amdgcn_wmma_f32_16x16x4_f32(false, a, false, bm,
                                                (short)0, acc, false, false);
  }
#pragma unroll
  for (int i = 0; i < 8; ++i)
    u[(size_t)(m0 + i + 8 * kh) * DIM + n0 + row] = acc[i];
}

// ---------------------------------------------------------------------------
// Kernel 2: flash pass over hx. One workgroup = (batch b, S-split).
// score[b,s] = (hx[b,s,:].u[b,:] + st[b].bias) * mask1 ; online softmax;
// accumulate v = sum attn-weighted hx rows. Partials (m, l, v) -> workspace.
// ---------------------------------------------------------------------------
__global__ __launch_bounds__(256) void k2_flash_partial(
    const float* __restrict__ st, const float* __restrict__ hx,
    const float* __restrict__ mask, const float* __restrict__ bias,
    const float* __restrict__ u,
    float* __restrict__ pm, float* __restrict__ pl, float* __restrict__ pv) {
  const int b     = blockIdx.x / SPLIT;
  const int split = blockIdx.x % SPLIT;
  const int tid   = threadIdx.x;
  const int lane  = tid & 31;
  const int w     = tid >> 5;

  __shared__ float red[256];
  __shared__ float wave_m[8], wave_l[8];
  __shared__ float vsh[8][DIM];

  // c[b] = st[b,:] . bias  (block reduction)
  float cpart = 0.f;
  for (int i = tid; i < DIM; i += 256)
    cpart += st[b * DIM + i] * bias[i];
  red[tid] = cpart;
  __syncthreads();
  for (int off = 128; off > 0; off >>= 1) {
    if (tid < off) red[tid] += red[tid + off];
    __syncthreads();
  }
  const float cb = red[0];

  // u[b,:] cached in registers: lane holds float4 chunks (32j + lane)
  float4 uv[8];
  const float4* u4 = (const float4*)(u + (size_t)b * DIM);
#pragma unroll
  for (int j = 0; j < 8; ++j) uv[j] = u4[32 * j + lane];

  float m_run = -__builtin_inff();
  float l_run = 0.f;
  float4 vacc[8];
#pragma unroll
  for (int j = 0; j < 8; ++j) { vacc[j].x = vacc[j].y = vacc[j].z = vacc[j].w = 0.f; }

  const int rows_per_split = SLEN / SPLIT;          // 128
  const int rows_per_wave  = rows_per_split / 8;    // 16
  const int s0 = split * rows_per_split + w * rows_per_wave;

  for (int r = 0; r < rows_per_wave; ++r) {
    const int s = s0 + r;
    const float* hrow = hx + ((size_t)b * SLEN + s) * DIM;
    // prefetch next row (4KB covered: 32 lanes * 128B) -> global_prefetch_b8
    if (r + 1 < rows_per_wave)
      __builtin_prefetch(hrow + DIM + lane * 32, 0, 0);

    const float4* h4 = (const float4*)hrow;
    float4 hv[8];
#pragma unroll
    for (int j = 0; j < 8; ++j) hv[j] = h4[32 * j + lane];

    float p = 0.f;
#pragma unroll
    for (int j = 0; j < 8; ++j)
      p += hv[j].x * uv[j].x + hv[j].y * uv[j].y +
           hv[j].z * uv[j].z + hv[j].w * uv[j].w;
    // wave32 reduction
    p += __shfl_xor(p, 16, 32);
    p += __shfl_xor(p,  8, 32);
    p += __shfl_xor(p,  4, 32);
    p += __shfl_xor(p,  2, 32);
    p += __shfl_xor(p,  1, 32);

    const float mk    = mask[b * SLEN + s];
    const float mk1   = mk + (1.f - mk) * 1e-18f;   // exact reference semantics
    const float score = (p + cb) * mk1;

    const float mn = fmaxf(m_run, score);
    const float sc = __expf(m_run - mn);
    const float pe = __expf(score - mn);
    l_run = l_run * sc + pe;
#pragma unroll
    for (int j = 0; j < 8; ++j) {
      vacc[j].x = vacc[j].x * sc + pe * hv[j].x;
      vacc[j].y = vacc[j].y * sc + pe * hv[j].y;
      vacc[j].z = vacc[j].z * sc + pe * hv[j].z;
      vacc[j].w = vacc[j].w * sc + pe * hv[j].w;
    }
    m_run = mn;
  }

  // merge the 8 per-wave partials inside the workgroup
  if (lane == 0) { wave_m[w] = m_run; wave_l[w] = l_run; }
  __syncthreads();
  float mb = wave_m[0];
#pragma unroll
  for (int i = 1; i < 8; ++i) mb = fmaxf(mb, wave_m[i]);
  const float f = __expf(m_run - mb);     // wave-uniform
#pragma unroll
  for (int j = 0; j < 8; ++j) {
    const int base = 4 * (32 * j + lane);
    vsh[w][base + 0] = vacc[j].x * f;
    vsh[w][base + 1] = vacc[j].y * f;
    vsh[w][base + 2] = vacc[j].z * f;
    vsh[w][base + 3] = vacc[j].w * f;
  }
  __syncthreads();

  float lb = 0.f;
#pragma unroll
  for (int i = 0; i < 8; ++i) lb += wave_l[i] * __expf(wave_m[i] - mb);

  const int pi = b * SPLIT + split;
  for (int d = tid; d < DIM; d += 256) {
    float sv = 0.f;
#pragma unroll
    for (int i = 0; i < 8; ++i) sv += vsh[i][d];
    pv[(size_t)pi * DIM + d] = sv;
  }
  if (tid == 0) { pm[pi] = mb; pl[pi] = lb; }
}

// ---------------------------------------------------------------------------
// Kernel 3: merge SPLIT partials per batch -> vnorm[b,d] = v_total / l_total
// ---------------------------------------------------------------------------
__global__ __launch_bounds__(256) void k3_combine(const float* __restrict__ pm,
                                                  const float* __restrict__ pl,
                                                  const float* __restrict__ pv,
                                                  float* __restrict__ vnorm) {
  const int b = blockIdx.x;
  const int tid = threadIdx.x;
  float mb = -__builtin_inff();
#pragma unroll
  for (int i = 0; i < SPLIT; ++i) mb = fmaxf(mb, pm[b * SPLIT + i]);
  float f[SPLIT];
  float L = 0.f;
#pragma unroll
  for (int i = 0; i < SPLIT; ++i) {
    f[i] = __expf(pm[b * SPLIT + i] - mb);
    L += pl[b * SPLIT + i] * f[i];
  }
  const float invL = 1.f / L;
  for (int d = tid; d < DIM; d += 256) {
    float sv = 0.f;
#pragma unroll
    for (int i = 0; i < SPLIT; ++i)
      sv += pv[(size_t)(b * SPLIT + i) * DIM + d] * f[i];
    vnorm[(size_t)b * DIM + d] = sv * invL;
  }
}

// ---------------------------------------------------------------------------
// Kernel 4: ct[b,d] = sum_e vnorm[b,e] * W[d,e] + bias[d]  (v @ W^T + bias)
// fp32 WMMA 16x16x4, B-matrix is W transposed (B[k=e, n=d] = W[d,e]).
// ---------------------------------------------------------------------------
__global__ __launch_bounds__(256) void k4_out_gemm(const float* __restrict__ vnorm,
                                                   const float* __restrict__ W,
                                                   const float* __restrict__ bias,
                                                   float* __restrict__ ct) {
  const int wave = (blockIdx.x * blockDim.x + threadIdx.x) >> 5;
  const int lane = threadIdx.x & 31;
  const int mt = wave >> 6;          // 0..1
  const int nt = wave & 63;          // 0..63
  const int row = lane & 15;
  const int kh  = lane >> 4;
  const int m0 = mt * 16, n0 = nt * 16;

  v8f acc = {};
  for (int k = 0; k < DIM; k += 4) {
    const int kk = k + 2 * kh;
    v2f a, bm;
    a.x  = vnorm[(size_t)(m0 + row) * DIM + kk];
    a.y  = vnorm[(size_t)(m0 + row) * DIM + kk + 1];
    bm.x = W[(size_t)(n0 + row) * DIM + kk];       // W^T access
    bm.y = W[(size_t)(n0 + row) * DIM + kk + 1];
    acc = __builtin_amdgcn_wmma_f32_16x16x4_f32(false, a, false, bm,
                                                (short)0, acc, false, false);
  }
  const float bd = bias[n0 + row];
#pragma unroll
  for (int i = 0; i < 8; ++i)
    ct[(size_t)(m0 + i + 8 * kh) * DIM + n0 + row] = acc[i] + bd;
}

// ---------------------------------------------------------------------------
extern "C" void kernel_launch(void* const* d_in, const int* in_sizes, int n_in,
                              void* d_out, int out_size, void* d_ws, size_t ws_size,
                              hipStream_t stream) {
  const float* st   = (const float*)d_in[0];   // [B, D]
  const float* hx   = (const float*)d_in[1];   // [B, S, D]
  const float* mask = (const float*)d_in[2];   // [B, S]
  const float* W    = (const float*)d_in[3];   // [D, D]
  const float* bias = (const float*)d_in[4];   // [D]
  float* out = (float*)d_out;                  // [B, D]

  // workspace layout (floats)
  float* wsf   = (float*)d_ws;
  float* u     = wsf;                                  // B*D        = 32768
  float* pm    = u  + (size_t)BATCH * DIM;             // B*SPLIT    = 512
  float* pl    = pm + (size_t)BATCH * SPLIT;           // B*SPLIT    = 512
  float* pv    = pl + (size_t)BATCH * SPLIT;           // B*SPLIT*D  = 524288
  float* vnorm = pv + (size_t)BATCH * SPLIT * DIM;     // B*D        = 32768

  // 1) u = st @ W  (WMMA fp32): 128 waves = 16 blocks x 256
  k1_u_gemm<<<16, 256, 0, stream>>>(st, W, u);

  // 2) flash pass over hx: B*SPLIT workgroups
  k2_flash_partial<<<BATCH * SPLIT, 256, 0, stream>>>(st, hx, mask, bias, u,
                                                      pm, pl, pv);

  // 3) merge splits -> normalized attention-weighted hx sum
  k3_combine<<<BATCH, 256, 0, stream>>>(pm, pl, pv, vnorm);

  // 4) ct = vnorm @ W^T + bias (WMMA fp32) -> d_out
  k4_out_gemm<<<16, 256, 0, stream>>>(vnorm, W, bias, out);
}